// Model_42606075576412
// MI455X (gfx1250) — compile-verified
//
#include <hip/hip_runtime.h>
#include <hip/hip_bf16.h>

// ---------------- model dims ----------------
#define BQ     8
#define SEQ    336
#define CN     862
#define CP     864              // C padded to multiple of 16
#define LDEC   1424
#define HD     16
#define FIN    32               // din + H
#define NPE    21
#define NPD    89
#define PRED   96
#define MROW   (BQ*CP)          // 6912 padded (b,node) rows
#define XCOL   (BQ*FIN)         // 256

typedef float v2f __attribute__((ext_vector_type(2)));
typedef float v8f __attribute__((ext_vector_type(8)));

static __device__ __forceinline__ v8f wmma4(v2f a, v2f b, v8f c) {
  // V_WMMA_F32_16X16X4_F32 : full fp32 matrix op (CDNA5)
  return __builtin_amdgcn_wmma_f32_16x16x4_f32(false, a, false, b, (short)0, c, false, false);
}

// ---------------- WMMA GEMMs (one wave = one 16x16 tile, wave32, EXEC all-1) ----

// D(M x N) = A(M x K, lda=K) @ B(K x N, ldb=N). grid=(M/16, N/16), block=32.
__global__ __launch_bounds__(32) void k_gemm_nn(const float* __restrict__ A,
    const float* __restrict__ B, float* __restrict__ D, int N, int K) {
  const int L = threadIdx.x, half = L >> 4, l16 = L & 15;
  const int arow = blockIdx.x * 16 + l16;
  const int bcol = blockIdx.y * 16 + l16;
  const float* ap = A + arow * K + half * 2;
  const float* bp = B + (half * 2) * N + bcol;
  v8f acc = {};
  for (int k0 = 0; k0 < K; k0 += 4) {
    v2f a; a.x = ap[k0];     a.y = ap[k0 + 1];
    v2f b; b.x = bp[k0 * N]; b.y = bp[k0 * N + N];
    acc = wmma4(a, b, acc);
  }
  const int r0 = blockIdx.x * 16 + half * 8;
  const int c0 = blockIdx.y * 16 + l16;
  #pragma unroll
  for (int v = 0; v < 8; ++v) D[(r0 + v) * N + c0] = acc[v];
}

// gate: ZR = sigmoid(xe @ gWp + gb[n]) ; K=128, N=32. grid=(MROW/16, 2)
__global__ __launch_bounds__(32) void k_gemm_gate(const float* __restrict__ XE,
    const float* __restrict__ W, float* __restrict__ ZR,
    const float* __restrict__ E, const float* __restrict__ gbp) {
  const int L = threadIdx.x, half = L >> 4, l16 = L & 15;
  const int arow = blockIdx.x * 16 + l16;
  const int o = blockIdx.y * 16 + l16;           // 0..31
  const float* ap = XE + arow * 128 + half * 2;
  const float* bp = W + (half * 2) * 32 + o;
  v8f acc = {};
  #pragma unroll 4
  for (int k0 = 0; k0 < 128; k0 += 4) {
    v2f a; a.x = ap[k0];      a.y = ap[k0 + 1];
    v2f b; b.x = bp[k0 * 32]; b.y = bp[k0 * 32 + 32];
    acc = wmma4(a, b, acc);
  }
  const int r0 = blockIdx.x * 16 + half * 8;
  #pragma unroll
  for (int v = 0; v < 8; ++v) {
    int row = r0 + v, n = row % CP;
    float bias = 0.f;
    if (n < CN) bias = E[n * 2] * gbp[o] + E[n * 2 + 1] * gbp[32 + o];
    float x = acc[v] + bias;
    ZR[row * 32 + o] = 1.f / (1.f + expf(-x));
  }
}

// candidate + fused GRU update: h = r*h + (1-r)*tanh(xe @ uWp + ub[n]); K=128,N=16
__global__ __launch_bounds__(32) void k_gemm_cand(const float* __restrict__ XE,
    const float* __restrict__ W, const float* __restrict__ ZR,
    float* __restrict__ Hl, const float* __restrict__ E, const float* __restrict__ ubp) {
  const int L = threadIdx.x, half = L >> 4, l16 = L & 15;
  const int arow = blockIdx.x * 16 + l16;
  const int o = l16;                              // N == 16
  const float* ap = XE + arow * 128 + half * 2;
  const float* bp = W + (half * 2) * 16 + o;
  v8f acc = {};
  #pragma unroll 4
  for (int k0 = 0; k0 < 128; k0 += 4) {
    v2f a; a.x = ap[k0];      a.y = ap[k0 + 1];
    v2f b; b.x = bp[k0 * 16]; b.y = bp[k0 * 16 + 16];
    acc = wmma4(a, b, acc);
  }
  const int r0 = blockIdx.x * 16 + half * 8;
  #pragma unroll
  for (int v = 0; v < 8; ++v) {
    int row = r0 + v, n = row % CP;
    float bias = 0.f;
    if (n < CN) bias = E[n * 2] * ubp[o] + E[n * 2 + 1] * ubp[16 + o];
    float hc = tanhf(acc[v] + bias);
    float r  = ZR[row * 32 + 16 + o];
    float h0 = Hl[row * 16 + o];
    Hl[row * 16 + o] = r * h0 + (1.f - r) * hc;
  }
}

// decoder projection rank-16 update: OACC += h1 @ Wproj[:, t*16:t*16+16]^T
// K=16, N=96. grid=(MROW/16, 6)
__global__ __launch_bounds__(32) void k_proj(const float* __restrict__ H1,
    const float* __restrict__ Wproj, int t, float* __restrict__ OACC) {
  const int L = threadIdx.x, half = L >> 4, l16 = L & 15;
  const int arow = blockIdx.x * 16 + l16;
  const int col = blockIdx.y * 16 + l16;
  const int r0 = blockIdx.x * 16 + half * 8;
  v8f acc;
  #pragma unroll
  for (int v = 0; v < 8; ++v) acc[v] = OACC[(r0 + v) * PRED + col];
  const float* ap = H1 + arow * 16 + half * 2;
  const float* bp = Wproj + col * (NPD * HD) + t * 16 + half * 2;  // B[k][col] = Wproj[col, t*16+k]
  #pragma unroll
  for (int k0 = 0; k0 < 16; k0 += 4) {
    v2f a; a.x = ap[k0]; a.y = ap[k0 + 1];
    v2f b; b.x = bp[k0]; b.y = bp[k0 + 1];
    acc = wmma4(a, b, acc);
  }
  #pragma unroll
  for (int v = 0; v < 8; ++v) OACC[(r0 + v) * PRED + col] = acc[v];
}

// ---------------- elementwise / setup kernels ----------------

// RevIN stats: mean & 1/std per (b, c)
__global__ void k_stats(const float* __restrict__ x, float* __restrict__ meanv,
                        float* __restrict__ istdv) {
  int idx = blockIdx.x * 256 + threadIdx.x;
  if (idx >= BQ * CP) return;
  int b = idx / CP, n = idx % CP;
  float mean = 0.f, istd = 1.f;
  if (n < CN) {
    float s = 0.f, ss = 0.f;
    for (int t = 0; t < SEQ; ++t) {
      float v = x[(b * SEQ + t) * CN + n];
      s += v; ss += v * v;
    }
    mean = s * (1.f / SEQ);
    float var = ss * (1.f / SEQ) - mean * mean;
    if (var < 0.f) var = 0.f;
    istd = rsqrtf(var + 1e-5f);
  }
  meanv[idx] = mean; istdv[idx] = istd;
}

// S = softmax(relu(E E^T), axis=1), zero-padded to 864x864. one block per row.
__global__ void k_supports(const float* __restrict__ E, float* __restrict__ S) {
  __shared__ float red[256];
  int n = blockIdx.x, tid = threadIdx.x;
  if (n >= CN) { for (int m = tid; m < CP; m += 256) S[n * CP + m] = 0.f; return; }
  float e0 = E[n * 2], e1 = E[n * 2 + 1];
  float lmax = -1e30f;
  for (int m = tid; m < CN; m += 256)
    lmax = fmaxf(lmax, fmaxf(0.f, e0 * E[m * 2] + e1 * E[m * 2 + 1]));
  red[tid] = lmax; __syncthreads();
  for (int s = 128; s > 0; s >>= 1) { if (tid < s) red[tid] = fmaxf(red[tid], red[tid + s]); __syncthreads(); }
  float mx = red[0]; __syncthreads();
  float lsum = 0.f;
  for (int m = tid; m < CN; m += 256)
    lsum += expf(fmaxf(0.f, e0 * E[m * 2] + e1 * E[m * 2 + 1]) - mx);
  red[tid] = lsum; __syncthreads();
  for (int s = 128; s > 0; s >>= 1) { if (tid < s) red[tid] += red[tid + s]; __syncthreads(); }
  float inv = 1.f / red[0];
  for (int m = tid; m < CP; m += 256) {
    float out = 0.f;
    if (m < CN) out = expf(fmaxf(0.f, e0 * E[m * 2] + e1 * E[m * 2 + 1]) - mx) * inv;
    S[n * CP + m] = out;
  }
}

// Build X (864 x 256): cols (b, j); j<16 -> layer input xt; j>=16 -> state h
// mode 0: trend enc (xt = revin_b[c]); 1: seasonal enc; 2: decoder (raw dec_inp); 3: layer-1 (xt = h0)
__global__ void k_build_xin(float* __restrict__ Xin, const float* __restrict__ src,
    int mode, int t, const float* __restrict__ meanv, const float* __restrict__ istdv,
    const float* __restrict__ rw, const float* __restrict__ rb,
    const float* __restrict__ hx, const float* __restrict__ hs) {
  int idx = blockIdx.x * 256 + threadIdx.x;
  if (idx >= CP * XCOL) return;
  int m = idx >> 8, col = idx & 255, b = col >> 5, j = col & 31;
  float val = 0.f;
  if (m < CN) {
    if (j < 16) {
      if (mode == 0)      val = rb[m];                                         // trend = time-mean of xn
      else if (mode == 1) val = (src[(b * SEQ + t * 16 + j) * CN + m] - meanv[b * CP + m])
                                * istdv[b * CP + m] * rw[m];                   // seasonal = xn - trend
      else if (mode == 2) val = src[(b * LDEC + t * 16 + j) * CN + m];
      else                val = hx[(b * CP + m) * 16 + j];
    } else {
      val = hs[(b * CP + m) * 16 + (j - 16)];
    }
  }
  Xin[m * XCOL + col] = val;
}

// Candidate input: j<16 copy xt from Xin ; j>=16 -> z * h
__global__ void k_build_xc(float* __restrict__ Xc, const float* __restrict__ Xin,
    const float* __restrict__ ZR, const float* __restrict__ hs) {
  int idx = blockIdx.x * 256 + threadIdx.x;
  if (idx >= CP * XCOL) return;
  int m = idx >> 8, col = idx & 255, b = col >> 5, j = col & 31;
  float val = 0.f;
  if (m < CN) {
    if (j < 16) val = Xin[m * XCOL + col];
    else {
      int row = b * CP + m;
      val = ZR[row * 32 + (j - 16)] * hs[row * 16 + (j - 16)];
    }
  }
  Xc[m * XCOL + col] = val;
}

// xe[row, (d,k,i)] = x_g[k][i] * E[n,d]  (k=0: X0=ins, k=1: X1=S@ins)
__global__ void k_build_xe(float* __restrict__ XE, const float* __restrict__ X0,
    const float* __restrict__ X1, const float* __restrict__ E) {
  int idx = blockIdx.x * 256 + threadIdx.x;
  if (idx >= MROW * 128) return;
  int row = idx >> 7, f = idx & 127;
  int d = f >> 6, k = (f >> 5) & 1, i = f & 31;
  int n = row % CP, b = row / CP;
  float val = 0.f;
  if (n < CN) {
    float s = (k == 0) ? X0[n * XCOL + b * 32 + i] : X1[n * XCOL + b * 32 + i];
    val = s * E[n * 2 + d];
  }
  XE[idx] = val;
}

// encoder->decoder init: h = h_enc @ Wd^T + bd (both layers; Hcopy is a snapshot)
__global__ void k_trans(float* __restrict__ H, const float* __restrict__ Hcopy,
                        const float* __restrict__ Wd, const float* __restrict__ bd) {
  int idx = blockIdx.x * 256 + threadIdx.x;
  if (idx >= 2 * MROW * 16) return;
  int l = idx / (MROW * 16), rem = idx % (MROW * 16), row = rem / 16, o = rem % 16;
  const float* hsrc = Hcopy + l * MROW * 16 + row * 16;
  float acc = bd[o];
  #pragma unroll
  for (int hh = 0; hh < 16; ++hh) acc += hsrc[hh] * Wd[o * 16 + hh];
  H[l * MROW * 16 + row * 16 + o] = acc;
}

// out[b,p,c] = ((acc + bproj_t[p] + bproj_s[p]) - rb[c]) / (rw[c]+1e-10) * std + mean
__global__ void k_final(const float* __restrict__ OACC, const float* __restrict__ rb,
    const float* __restrict__ rw, const float* __restrict__ meanv,
    const float* __restrict__ istdv, const float* __restrict__ bpt,
    const float* __restrict__ bps, float* __restrict__ out) {
  int idx = blockIdx.x * 256 + threadIdx.x;
  if (idx >= BQ * PRED * CN) return;
  int b = idx / (PRED * CN), rem = idx % (PRED * CN), p = rem / CN, c = rem % CN;
  float acc = OACC[(b * CP + c) * PRED + p] + bpt[p] + bps[p];
  float t1 = (acc - rb[c]) / (rw[c] + 1e-10f);
  float stdv = 1.f / istdv[b * CP + c];
  out[idx] = t1 * stdv + meanv[b * CP + c];
}

// ---------------- host orchestration ----------------

struct LayerW { const float *gWp, *gbp, *uWp, *ubp; };
struct BackboneW { const float *E, *Wd, *Wproj, *bd, *bproj; LayerW dec[2], enc[2]; };

extern "C" void kernel_launch(void* const* d_in, const int* in_sizes, int n_in,
                              void* d_out, int out_size, void* d_ws, size_t ws_size,
                              hipStream_t stream) {
  // jax tree_flatten (sorted dict keys) leaf order:
  // 0: dec_inp, 1: revin_b, 2: revin_w, 3..23: seasonal, 24..44: trend, 45: x
  // per backbone: E, Wd, Wproj, bd, bproj, then dec[0..1]/enc[0..1] each {gWp,gbp,uWp,ubp}
  const float* dec_inp = (const float*)d_in[0];
  const float* revin_b = (const float*)d_in[1];
  const float* revin_w = (const float*)d_in[2];
  const float* x       = (const float*)d_in[45];

  auto getBB = [&](int base, BackboneW& bb) {
    bb.E = (const float*)d_in[base + 0];
    bb.Wd = (const float*)d_in[base + 1];
    bb.Wproj = (const float*)d_in[base + 2];
    bb.bd = (const float*)d_in[base + 3];
    bb.bproj = (const float*)d_in[base + 4];
    int o = base + 5;
    for (int s = 0; s < 2; ++s)
      for (int l = 0; l < 2; ++l) {
        LayerW& L = (s == 0) ? bb.dec[l] : bb.enc[l];
        L.gWp = (const float*)d_in[o + 0];
        L.gbp = (const float*)d_in[o + 1];
        L.uWp = (const float*)d_in[o + 2];
        L.ubp = (const float*)d_in[o + 3];
        o += 4;
      }
  };
  BackboneW seasBB, trendBB;
  getBB(3, seasBB);
  getBB(24, trendBB);

  // workspace layout (floats)
  float* w = (float*)d_ws;
  size_t off = 0;
  auto alloc = [&](size_t n) { float* p = w + off; off += n; return p; };
  float* meanv = alloc(BQ * CP);
  float* istdv = alloc(BQ * CP);
  float* Spad  = alloc((size_t)CP * CP);
  float* Xin   = alloc((size_t)CP * XCOL);
  float* SX    = alloc((size_t)CP * XCOL);
  float* Xc    = alloc((size_t)CP * XCOL);
  float* SXc   = alloc((size_t)CP * XCOL);
  float* XE    = alloc((size_t)MROW * 128);
  float* ZR    = alloc((size_t)MROW * 32);
  float* H     = alloc((size_t)2 * MROW * 16);
  float* Hcopy = alloc((size_t)2 * MROW * 16);
  float* OACC  = alloc((size_t)MROW * PRED);
  if (off * sizeof(float) > ws_size) return;  // insufficient scratch (deterministic no-op)

  const dim3 blk32(32), blk256(256);
  const dim3 g1(CP / 16, XCOL / 16);     // 54 x 16  (S @ X)
  const dim3 g2g(MROW / 16, 2);          // gate GEMM
  const dim3 g2c(MROW / 16, 1);          // cand GEMM
  const dim3 gpr(MROW / 16, PRED / 16);  // projection update
  const int nbXin = (CP * XCOL + 255) / 256;
  const int nbXe  = (MROW * 128 + 255) / 256;

  k_stats<<<(BQ * CP + 255) / 256, blk256, 0, stream>>>(x, meanv, istdv);
  hipMemsetAsync(OACC, 0, (size_t)MROW * PRED * sizeof(float), stream);

  auto run_backbone = [&](const BackboneW& bb, int enc_mode) {
    k_supports<<<CP, blk256, 0, stream>>>(bb.E, Spad);
    hipMemsetAsync(H, 0, (size_t)2 * MROW * 16 * sizeof(float), stream);
    for (int stage = 0; stage < 2; ++stage) {
      const int steps = (stage == 0) ? NPE : NPD;
      const LayerW* Ls = (stage == 0) ? bb.enc : bb.dec;
      const float* src = (stage == 0) ? x : dec_inp;
      for (int t = 0; t < steps; ++t) {
        for (int l = 0; l < 2; ++l) {
          float* Hl = H + (size_t)l * MROW * 16;
          int mode = (l == 1) ? 3 : ((stage == 0) ? enc_mode : 2);
          k_build_xin<<<nbXin, blk256, 0, stream>>>(Xin, src, mode, t, meanv, istdv,
                                                    revin_w, revin_b, H, Hl);
          k_gemm_nn<<<g1, blk32, 0, stream>>>(Spad, Xin, SX, XCOL, CP);
          k_build_xe<<<nbXe, blk256, 0, stream>>>(XE, Xin, SX, bb.E);
          k_gemm_gate<<<g2g, blk32, 0, stream>>>(XE, Ls[l].gWp, ZR, bb.E, Ls[l].gbp);
          k_build_xc<<<nbXin, blk256, 0, stream>>>(Xc, Xin, ZR, Hl);
          k_gemm_nn<<<g1, blk32, 0, stream>>>(Spad, Xc, SXc, XCOL, CP);
          k_build_xe<<<nbXe, blk256, 0, stream>>>(XE, Xc, SXc, bb.E);
          k_gemm_cand<<<g2c, blk32, 0, stream>>>(XE, Ls[l].uWp, ZR, Hl, bb.E, Ls[l].ubp);
        }
        if (stage == 1)
          k_proj<<<gpr, blk32, 0, stream>>>(H + (size_t)MROW * 16, bb.Wproj, t, OACC);
      }
      if (stage == 0) {
        hipMemcpyAsync(Hcopy, H, (size_t)2 * MROW * 16 * sizeof(float),
                       hipMemcpyDeviceToDevice, stream);
        k_trans<<<(2 * MROW * 16 + 255) / 256, blk256, 0, stream>>>(H, Hcopy, bb.Wd, bb.bd);
      }
    }
  };

  run_backbone(trendBB, /*enc_mode=*/0);
  run_backbone(seasBB,  /*enc_mode=*/1);

  k_final<<<(BQ * PRED * CN + 255) / 256, blk256, 0, stream>>>(
      OACC, revin_b, revin_w, meanv, istdv, trendBB.bproj, seasBB.bproj, (float*)d_out);
}